// JSCCQ_31550829757033
// MI455X (gfx1250) — compile-verified
//
#include <hip/hip_runtime.h>
#include <hip/hip_bf16.h>

typedef __attribute__((ext_vector_type(2))) float v2f;
typedef __attribute__((ext_vector_type(8))) float v8f;

#define SIGMA 10.0f
#define LOG2E 1.44269504088896340736f
// sigma pre-scaled by log2(e): softmax computed with raw v_exp_f32 (exp2),
// exactly equivalent to exp(-SIGMA*dist) softmax (scale-invariant shift by mx).
#define SIGMA2E (SIGMA * LOG2E)

// Zero the likelihoods slice of d_out (atomically accumulated by main kernel).
__global__ void vq_zero_lik(float* lik) {
    if (threadIdx.x < 64) lik[threadIdx.x] = 0.0f;
}

// Transposed mapping: A (16x4 f32, loop-invariant) holds codeword affine rows
// (e_x, e_y, -sigma'*|e|^2, 0); B (4x16 f32, per tile) holds point columns
// (2*sigma'*p_x, 2*sigma'*p_y, 1, 0).  D[m][n] = score'(codeword m, point n).
// Each point's 64 scores live in exactly 2 lanes (n, n+16): softmax reductions
// are in-register trees + ONE xor-16 shuffle each.
__launch_bounds__(256)
__global__ void vq_kernel(const float* __restrict__ x,
                          const float* __restrict__ embed,
                          float* __restrict__ quant,   // [N,2] flat
                          float* __restrict__ lik,     // [64]
                          int numTiles, float invN)
{
    __shared__ float slik[64];

    const int lane       = threadIdx.x & 31;
    const int laneInHalf = lane & 15;
    const int halfId     = lane >> 4;
    const int wave       = (int)((blockIdx.x * blockDim.x + threadIdx.x) >> 5);
    const int totalWaves = (int)((gridDim.x * blockDim.x) >> 5);

    if (threadIdx.x < 64) slik[threadIdx.x] = 0.0f;
    __syncthreads();

    // A operands: 4 codeword tiles, loaded once.
    // f32 16x4 A layout: lane L<16 -> M=L, (K0,K1); lane L>=16 -> M=L-16, (K2,K3).
    v2f a[4];
    #pragma unroll
    for (int t = 0; t < 4; ++t) {
        const int j = 16 * t + laneInHalf;
        const float ex = embed[2 * j];
        const float ey = embed[2 * j + 1];
        if (halfId == 0) { a[t].x = ex;                              a[t].y = ey;   }
        else             { a[t].x = -SIGMA2E * (ex * ex + ey * ey);  a[t].y = 0.0f; }
    }

    // Per-lane likelihood accumulators: reg i <-> codeword 16*(i/8) + 8*halfId + i%8
    float acc[32];
    #pragma unroll
    for (int i = 0; i < 32; ++i) acc[i] = 0.0f;

    const float2* x2 = (const float2*)x;
    float2* quant2 = (float2*)quant;
    const bool hiHalf = (halfId != 0);

    for (int tile = wave; tile < numTiles; tile += totalWaves) {
        const int base = tile * 16;

        // prefetch next tile's points (global_prefetch_b8, speculative)
        __builtin_prefetch((const void*)(x2 + base + totalWaves * 16), 0, 1);

        // B operand: f32 4x16 layout: lane n<16 -> (K0,K1) col n; lane>=16 -> (K2,K3).
        // lanes 16-31 alias lanes 0-15's addresses, so load unconditionally (no
        // exec-masked branch) and select: keeps EXEC all-ones straight into WMMA.
        const float2 p = x2[base + laneInHalf];
        v2f b;
        b.x = hiHalf ? 1.0f : 2.0f * SIGMA2E * p.x;  // K2 = 1 picks up -sigma'*|e|^2
        b.y = hiHalf ? 0.0f : 2.0f * SIGMA2E * p.y;  // K3 unused

        v8f c = {};
        // 8 args: (neg_a, A, neg_b, B, c_mod, C, reuse_a, reuse_b)
        v8f d0 = __builtin_amdgcn_wmma_f32_16x16x4_f32(false, a[0], false, b, (short)0, c, false, false);
        v8f d1 = __builtin_amdgcn_wmma_f32_16x16x4_f32(false, a[1], false, b, (short)0, c, false, false);
        v8f d2 = __builtin_amdgcn_wmma_f32_16x16x4_f32(false, a[2], false, b, (short)0, c, false, false);
        v8f d3 = __builtin_amdgcn_wmma_f32_16x16x4_f32(false, a[3], false, b, (short)0, c, false, false);

        // This lane's 32 scores: s[i], codeword j(i) = 16*(i/8)+8*halfId+(i%8),
        // point = base + (lane&15). j(i) strictly increasing in i.
        float s[32];
        #pragma unroll
        for (int v = 0; v < 8; ++v) {
            s[v]      = d0[v];
            s[8 + v]  = d1[v];
            s[16 + v] = d2[v];
            s[24 + v] = d3[v];
        }

        // max over 64 scores: in-register tree + one cross-half exchange
        float mx = fmaxf(s[0], s[1]);
        #pragma unroll
        for (int i = 2; i < 32; ++i) mx = fmaxf(mx, s[i]);
        mx = fmaxf(mx, __shfl_xor(mx, 16, 32));

        // raw v_exp_f32 numerators (terms below 2^-126 are negligible in the sum,
        // matching f32 reference underflow) + 4-way ILP denominator reduction
        float e[32];
        #pragma unroll
        for (int i = 0; i < 32; ++i) e[i] = __builtin_amdgcn_exp2f(s[i] - mx);

        float p0 = 0.0f, p1 = 0.0f, p2 = 0.0f, p3 = 0.0f;
        #pragma unroll
        for (int i = 0; i < 32; i += 4) {
            p0 += e[i];
            p1 += e[i + 1];
            p2 += e[i + 2];
            p3 += e[i + 3];
        }
        float sum = (p0 + p1) + (p2 + p3);
        sum += __shfl_xor(sum, 16, 32);
        const float r = __builtin_amdgcn_rcpf(sum);

        #pragma unroll
        for (int i = 0; i < 32; ++i) acc[i] += e[i] * r;

        // argmax with first-index (smallest j) tie-break: descending scan, j(i)
        // monotone in i, then cross-half min.
        unsigned cand = 0xFFFFu;
        #pragma unroll
        for (int i = 31; i >= 0; --i) {
            const unsigned j = 16u * (unsigned)(i >> 3) + 8u * (unsigned)halfId + (unsigned)(i & 7);
            if (s[i] == mx) cand = j;
        }
        cand = min(cand, (unsigned)__shfl_xor((int)cand, 16, 32));

        // hard quantize (straight-through forward): contiguous float2 store
        if (halfId == 0) {
            float2 q;
            q.x = embed[2 * cand];
            q.y = embed[2 * cand + 1];
            quant2[base + laneInHalf] = q;
        }
    }

    // Reduce acc across the 16 lanes of each half (once per wave), then
    // block-level LDS combine, then one global atomic per codeword per block.
    #pragma unroll
    for (int i = 0; i < 32; ++i) {
        acc[i] += __shfl_xor(acc[i], 8, 32);
        acc[i] += __shfl_xor(acc[i], 4, 32);
        acc[i] += __shfl_xor(acc[i], 2, 32);
        acc[i] += __shfl_xor(acc[i], 1, 32);
    }
    if (laneInHalf == 0) {
        #pragma unroll
        for (int i = 0; i < 32; ++i) {
            const int j = 16 * (i >> 3) + 8 * halfId + (i & 7);
            atomicAdd(&slik[j], acc[i]);   // ds_add_f32
        }
    }
    __syncthreads();
    if (threadIdx.x < 64) atomicAdd(&lik[threadIdx.x], slik[threadIdx.x] * invN);
}

extern "C" void kernel_launch(void* const* d_in, const int* in_sizes, int n_in,
                              void* d_out, int out_size, void* d_ws, size_t ws_size,
                              hipStream_t stream) {
    (void)n_in; (void)d_ws; (void)ws_size; (void)out_size;

    const float* x     = (const float*)d_in[0];   // [64,32,32,32] f32
    const float* embed = (const float*)d_in[1];   // [64,2] f32
    float* quant = (float*)d_out;                 // [N,2] = in_sizes[0] floats
    float* lik   = (float*)d_out + in_sizes[0];   // [64]

    const int N        = in_sizes[0] / 2;   // 1,048,576 points
    const int numTiles = N / 16;            // 65,536 tiles
    const float invN   = 1.0f / (float)N;

    vq_zero_lik<<<1, 64, 0, stream>>>(lik);

    const int threads = 256;                // 8 waves/block (wave32)
    const int blocks  = 1024;               // 8192 waves -> 8 tiles/wave, uniform
    vq_kernel<<<blocks, threads, 0, stream>>>(x, embed, quant, lik, numTiles, invN);
}